// CentroidLayer_68023692034068
// MI455X (gfx1250) — compile-verified
//
#include <hip/hip_runtime.h>
#include <hip/hip_bf16.h>

typedef __attribute__((ext_vector_type(2))) float v2f;
typedef __attribute__((ext_vector_type(8))) float v8f;
typedef __attribute__((ext_vector_type(4))) unsigned int u32x4;
typedef __attribute__((ext_vector_type(8))) int i32x8;
typedef __attribute__((ext_vector_type(4))) int i32x4;

constexpr int P = 1024;   // n_prototypes
constexpr int D = 512;    // embed_dim
constexpr int B = 65536;  // batch

constexpr int WAVES          = 4;            // waves per block
constexpr int ROWS_PER_BLOCK = WAVES * 16;   // 64 queries per block
// TDM pads 2 DWORDs after every 256 DWORDs stored -> effective row stride of
// 512-float rows is 516 floats (mid-row pad at k=256, end pad at k=512).
// 516 % 64 == 4 -> conflict-free ds_load_b64 operand fetches.
constexpr int STRIDE   = 516;
constexpr int NT2      = P / 32;             // 32 iterations of 32 centroids
constexpr int B_ROWS   = 32;                 // centroid rows per staged tile

// ---------------------------------------------------------------------------
// TDM: one-instruction DMA of a 2D f32 tile (global, row-major) into LDS,
// with pad_interval=256 DW / pad_amount=2 DW to produce the 516-float stride.
// Issued per-wave (EXEC ignored); tracked by TENSORcnt.
// ---------------------------------------------------------------------------
__device__ __forceinline__ void tdm_load_2d_f32(void* lds_dst,
                                                const void* gsrc,
                                                unsigned int tile_d0,      // elems per row
                                                unsigned int tile_d1,      // rows
                                                unsigned int row_stride) { // elems
    const unsigned long long ga = (unsigned long long)gsrc;
    const unsigned int lds_off  = (unsigned int)(unsigned long long)lds_dst; // AS3 offset

    u32x4 g0;
    g0.x = 1u;                                       // count=1 (valid user D#)
    g0.y = lds_off;                                  // lds_addr (bytes)
    g0.z = (unsigned int)(ga & 0xffffffffull);       // global_addr[31:0]
    g0.w = (unsigned int)((ga >> 32) & 0x1ffffffull) // global_addr[56:32]
         | (2u << 30);                               // type = 2 ("image")

    i32x8 g1 = {};
    g1[0] = (int)((2u << 16)      // data_size = 4 bytes
                | (1u << 20)      // pad_enable
                | (7u << 22)      // pad_interval: 256 DWORDs
                | (1u << 25));    // pad_amount:   2 DWORDs
    g1[1] = (int)((tile_d0 & 0xffffu) << 16);                                  // tensor_dim0[15:0]
    g1[2] = (int)(((tile_d0 >> 16) & 0xffffu) | ((tile_d1 & 0xffffu) << 16));  // t_dim0 hi | tensor_dim1 lo
    g1[3] = (int)(((tile_d1 >> 16) & 0xffffu) | ((tile_d0 & 0xffffu) << 16));  // t_dim1 hi | tile_dim0
    g1[4] = (int)(tile_d1 & 0xffffu);                                          // tile_dim1 (tile_dim2 = 0)
    g1[5] = (int)row_stride;                                                   // tensor_dim0_stride lo32
    g1[6] = 0;
    g1[7] = 0;

    i32x4 gz = {};
#if __clang_major__ >= 23
    i32x8 gz8 = {};
    __builtin_amdgcn_tensor_load_to_lds(g0, g1, gz, gz, gz8, 0);
#else
    __builtin_amdgcn_tensor_load_to_lds(g0, g1, gz, gz, 0);
#endif
}

// ---------------------------------------------------------------------------
// Kernel 1: normalize centroid rows (c / max(||c||, eps)) into workspace
// one wave32 per row; 512 floats -> 4 float4 per lane
// ---------------------------------------------------------------------------
__global__ void centroid_normalize_kernel(const float* __restrict__ cent,
                                          float* __restrict__ cnorm) {
    const int wave = (blockIdx.x * blockDim.x + threadIdx.x) >> 5;
    const int lane = threadIdx.x & 31;
    if (wave >= P) return;

    const float4* row = reinterpret_cast<const float4*>(cent + (size_t)wave * D);
    float4 v[4];
    float ssq = 0.0f;
#pragma unroll
    for (int i = 0; i < 4; ++i) {
        v[i] = row[i * 32 + lane];
        ssq += v[i].x * v[i].x + v[i].y * v[i].y + v[i].z * v[i].z + v[i].w * v[i].w;
    }
#pragma unroll
    for (int off = 16; off > 0; off >>= 1)
        ssq += __shfl_xor(ssq, off, 32);

    const float inv = 1.0f / fmaxf(sqrtf(ssq), 1e-12f);

    float4* orow = reinterpret_cast<float4*>(cnorm + (size_t)wave * D);
#pragma unroll
    for (int i = 0; i < 4; ++i) {
        float4 o = v[i];
        o.x *= inv; o.y *= inv; o.z *= inv; o.w *= inv;
        orow[i * 32 + lane] = o;
    }
}

// ---------------------------------------------------------------------------
// Kernel 2: fused logits-GEMM (f32 WMMA) + argmax + gather + one-hot epilogue
//   block: 128 threads = 4 waves, 64 queries (16 per wave)
//   A (query tile) stationary in LDS via one TDM load; B (32 normalized
//   centroids x 512) double-buffered via TDM, overlapped with compute.
//   Two independent accumulator chains per wave (two 16-col tiles).
// ---------------------------------------------------------------------------
__global__ void centroid_route_kernel(const float* __restrict__ q,
                                      const float* __restrict__ cent,    // original centroids (gather)
                                      const float* __restrict__ cnorm,   // normalized centroids
                                      float* __restrict__ ctx,           // (B, D)
                                      float* __restrict__ assign_out,    // (B,)
                                      float* __restrict__ rout)          // (B, P)
{
    extern __shared__ float smem[];
    float* sA   = smem;                                     // 64 x STRIDE
    float* sB   = smem + ROWS_PER_BLOCK * STRIDE;           // 2 x 32 x STRIDE
    int*   sIdx = reinterpret_cast<int*>(sB + 2 * B_ROWS * STRIDE); // 64 ints

    const int tid       = threadIdx.x;
    const int lane      = tid & 31;
    const int wave      = tid >> 5;
    const int blockRow0 = blockIdx.x * ROWS_PER_BLOCK;

    // ---- prologue: DMA the query tile + first centroid tile (wave 0) ----
    if (wave == 0) {
        tdm_load_2d_f32(sA, q + (size_t)blockRow0 * D, D, ROWS_PER_BLOCK, D);
        tdm_load_2d_f32(sB, cnorm, D, B_ROWS, D);
    }

    float bestVal[8];
    int   bestIdx[8];
#pragma unroll
    for (int r = 0; r < 8; ++r) { bestVal[r] = -3.0e38f; bestIdx[r] = 0; }

    const int mBase = wave * 16;                 // this wave's query rows
    const int aRow  = mBase + (lane & 15);
    const int kHalf = (lane >> 4) * 2;           // lanes 16..31 hold K+2,K+3
    const float* aBase  = sA + aRow * STRIDE + kHalf;
    const float* bBase  = sB + (lane & 15) * STRIDE + kHalf;

    for (int j = 0; j < NT2; ++j) {
        if (wave == 0)
            __builtin_amdgcn_s_wait_tensorcnt(0);   // buffer j&1 fully written
        __syncthreads();                            // visible to all waves

        // DMA next tile into the alternate buffer while computing this one
        if (wave == 0 && j + 1 < NT2)
            tdm_load_2d_f32(sB + ((j + 1) & 1) * B_ROWS * STRIDE,
                            cnorm + (size_t)(j + 1) * B_ROWS * D,
                            D, B_ROWS, D);

        const float* b0 = bBase + (j & 1) * B_ROWS * STRIDE;
        const float* b1 = b0 + 16 * STRIDE;

        v8f acc0 = {};
        v8f acc1 = {};
        // K split at 256 because the TDM pad adds +2 floats mid-row
#pragma unroll
        for (int half = 0; half < 2; ++half) {
            const int o = half * 258;               // 256 elems + 2 pad
#pragma unroll 8
            for (int k0 = 0; k0 < 256; k0 += 4) {
                v2f a  = *reinterpret_cast<const v2f*>(aBase + o + k0);
                v2f x0 = *reinterpret_cast<const v2f*>(b0 + o + k0);
                v2f x1 = *reinterpret_cast<const v2f*>(b1 + o + k0);
                acc0 = __builtin_amdgcn_wmma_f32_16x16x4_f32(
                    false, a, false, x0, (short)0, acc0, false, false);
                acc1 = __builtin_amdgcn_wmma_f32_16x16x4_f32(
                    false, a, false, x1, (short)0, acc1, false, false);
            }
        }

        // ---- running argmax (strict >, ascending cols => first-index ties) ----
        const int c0 = j * 32 + (lane & 15);
#pragma unroll
        for (int r = 0; r < 8; ++r) {
            if (acc0[r] > bestVal[r]) { bestVal[r] = acc0[r]; bestIdx[r] = c0; }
            if (acc1[r] > bestVal[r]) { bestVal[r] = acc1[r]; bestIdx[r] = c0 + 16; }
        }
    }

    // ---- reduce argmax across the 16 lanes of each half (wave32) ----
#pragma unroll
    for (int r = 0; r < 8; ++r) {
        float v = bestVal[r];
        int   i = bestIdx[r];
#pragma unroll
        for (int off = 1; off < 16; off <<= 1) {
            const float ov = __shfl_xor(v, off, 32);
            const int   oi = __shfl_xor(i, off, 32);
            if (ov > v || (ov == v && oi < i)) { v = ov; i = oi; }
        }
        bestIdx[r] = i;
    }
    if ((lane & 15) == 0) {
#pragma unroll
        for (int r = 0; r < 8; ++r)
            sIdx[mBase + (lane >> 4) * 8 + r] = bestIdx[r];
    }
    __syncthreads();

    // ---- fused epilogue: assignment, centroid gather, one-hot row ----
    for (int r2 = 0; r2 < 16; ++r2) {
        const int row = mBase + r2;
        const int b   = blockRow0 + row;
        const int idx = sIdx[row];

        if (lane == 0) assign_out[b] = (float)idx;

        // context_emb[b] = centroid_emb[idx]  (unnormalized)
        const float4* src = reinterpret_cast<const float4*>(cent + (size_t)idx * D);
        float4* dst = reinterpret_cast<float4*>(ctx + (size_t)b * D);
#pragma unroll
        for (int c4 = 0; c4 < 4; ++c4)
            dst[c4 * 32 + lane] = src[c4 * 32 + lane];

        // routing_probs[b] = one_hot(idx, P)
        float4* rdst = reinterpret_cast<float4*>(rout + (size_t)b * P);
#pragma unroll
        for (int c4 = 0; c4 < 8; ++c4) {
            float4 z = {0.0f, 0.0f, 0.0f, 0.0f};
            const int base = (c4 * 32 + lane) * 4;
            if (idx >= base && idx < base + 4)
                (&z.x)[idx - base] = 1.0f;
            rdst[c4 * 32 + lane] = z;
        }
    }
}

// ---------------------------------------------------------------------------
extern "C" void kernel_launch(void* const* d_in, const int* in_sizes, int n_in,
                              void* d_out, int out_size, void* d_ws, size_t ws_size,
                              hipStream_t stream) {
    const float* q    = (const float*)d_in[0];   // (B, D)
    const float* cent = (const float*)d_in[1];   // (P, D)

    float* out        = (float*)d_out;
    float* ctx        = out;                          // B*D
    float* assign_out = out + (size_t)B * D;          // B
    float* rout       = assign_out + B;               // B*P

    float* cnorm = (float*)d_ws;                      // P*D floats (2 MB)

    centroid_normalize_kernel<<<P / 8, 256, 0, stream>>>(cent, cnorm);

    const size_t smemBytes =
        (size_t)(ROWS_PER_BLOCK * STRIDE + 2 * B_ROWS * STRIDE) * sizeof(float) +
        (size_t)ROWS_PER_BLOCK * sizeof(int);

    centroid_route_kernel<<<B / ROWS_PER_BLOCK, WAVES * 32, smemBytes, stream>>>(
        q, cent, cnorm, ctx, assign_out, rout);
}